// S_GCN_51032801411524
// MI455X (gfx1250) — compile-verified
//
#include <hip/hip_runtime.h>
#include <hip/hip_bf16.h>
#include <math.h>

#define NN 50000
#define EE 1600000
#define DD 256

typedef __attribute__((ext_vector_type(2))) float v2f;
typedef __attribute__((ext_vector_type(8))) float v8f;
typedef __attribute__((ext_vector_type(4))) unsigned int v4u;
typedef __attribute__((ext_vector_type(4))) int v4i;
typedef __attribute__((ext_vector_type(8))) int v8i;

// ---------------- degree / norm ----------------
__global__ __launch_bounds__(256) void k_deg_init(float* __restrict__ deg) {
    int i = blockIdx.x * 256 + threadIdx.x;
    if (i < NN) deg[i] = 1.0f;   // self-loop contributes 1
}

__global__ __launch_bounds__(256) void k_deg_count(const int* __restrict__ col,
                                                   float* __restrict__ deg) {
    int e = blockIdx.x * 256 + threadIdx.x;
    if (e < EE) atomicAdd(&deg[col[e]], 1.0f);
}

__global__ __launch_bounds__(256) void k_deg_rsqrt(float* __restrict__ deg) {
    int i = blockIdx.x * 256 + threadIdx.x;
    if (i < NN) deg[i] = rsqrtf(deg[i]);   // deg >= 1 always (self-loops)
}

// ---------------- h = x @ W via V_WMMA_F32_16X16X4_F32 + TDM staging ----------
// grid: (3125, 4), block: 128 (4 waves). The x row-panel (16x256 fp32) is DMA'd
// into LDS by the Tensor Data Mover with pad_interval/pad_amount reproducing the
// bank-conflict-free 260-float row stride. Wave w computes tile (m0, n0).
__global__ __launch_bounds__(128) void k_gemm(const float* __restrict__ x,
                                              const float* __restrict__ W,
                                              float* __restrict__ h) {
    __shared__ float As[16][260];   // row stride 260: 4*m mod 64 distinct -> conflict-free

    const int m0   = blockIdx.x * 16;
    const int tid  = threadIdx.x;
    const int wave = tid >> 5;
    const int lane = tid & 31;

    if (wave == 0) {
        // ---- Tensor DMA descriptor (D#): 2-D tile, global -> LDS with padding ----
        unsigned lds_base = (unsigned)(unsigned long long)(&As[0][0]);
        unsigned long long ga = (unsigned long long)(const void*)(x + (size_t)m0 * DD);

        v4u g0;
        g0.x = 1u;                                   // count=1 (valid), user mode
        g0.y = lds_base;                             // lds_addr (bytes)
        g0.z = (unsigned)(ga & 0xFFFFFFFFu);         // global_addr[31:0]
        g0.w = (unsigned)((ga >> 32) & 0x01FFFFFFu)  // global_addr[56:32]
             | 0x80000000u;                          // type=2 ("image")

        v8i g1;
        g1[0] = (int)((2u << 16)       // data_size = 2 -> 4 bytes
                    | (1u << 20)       // pad_enable
                    | (7u << 22)       // pad_interval: pad after 256 DWORDs (one row)
                    | (3u << 25));     // pad_amount: 4 DWORDs (16B) -> stride 260 floats
        g1[1] = (int)(256u << 16);     // tensor_dim0 = 256 elements
        g1[2] = (int)(16u  << 16);     // tensor_dim1 = 16 rows
        g1[3] = (int)(256u << 16);     // tile_dim0 = 256
        g1[4] = 16;                    // tile_dim1 = 16
        g1[5] = 256;                   // tensor_dim0_stride = 256 elements
        g1[6] = 0;
        g1[7] = 0;

        v4i gz4 = {0, 0, 0, 0};        // groups 2/3 unused (2-D tensor)
        v8i gz8 = {0, 0, 0, 0, 0, 0, 0, 0};
        __builtin_amdgcn_tensor_load_to_lds(g0, g1, gz4, gz4, gz8, 0);
        __builtin_amdgcn_s_wait_tensorcnt(0);        // TENSORcnt -> 0 before barrier
    }
    __syncthreads();

    const int g  = lane >> 4;    // half-wave: K sub-block select
    const int ml = lane & 15;    // A: row M ; B/C/D: column N
    const int n  = blockIdx.y * 64 + wave * 16 + ml;

    v8f acc = {};
    #pragma unroll 8
    for (int kb = 0; kb < DD; kb += 4) {
        const int k = kb + g * 2;
        v2f a;
        a.x = As[ml][k];
        a.y = As[ml][k + 1];
        v2f bfrag;
        bfrag.x = W[(size_t)k * DD + n];
        bfrag.y = W[(size_t)(k + 1) * DD + n];
        acc = __builtin_amdgcn_wmma_f32_16x16x4_f32(
            /*neg_a=*/false, a, /*neg_b=*/false, bfrag,
            /*c_mod=*/(short)0, acc, /*reuse_a=*/false, /*reuse_b=*/false);
    }

    // C/D layout: lanes 0-15 -> M = r, lanes 16-31 -> M = r + 8; N = lane&15
    #pragma unroll
    for (int r = 0; r < 8; ++r) {
        h[(size_t)(m0 + g * 8 + r) * DD + n] = acc[r];
    }
}

// ---------------- out = b + self-loop term ----------------
__global__ __launch_bounds__(256) void k_init_out(const float* __restrict__ h,
                                                  const float* __restrict__ dis,
                                                  const float* __restrict__ bias,
                                                  float* __restrict__ out) {
    const int i = blockIdx.x;
    const int d = threadIdx.x;
    const float s = dis[i];
    out[(size_t)i * DD + d] = bias[d] + h[(size_t)i * DD + d] * s * s;
}

// ---------------- edge scatter-add (L2-resident atomics) ----------------
// 64 threads per edge, 4 floats per thread.
__global__ __launch_bounds__(256) void k_scatter(const int* __restrict__ row,
                                                 const int* __restrict__ col,
                                                 const float* __restrict__ h,
                                                 const float* __restrict__ dis,
                                                 float* __restrict__ out) {
    const int e = blockIdx.x * 4 + (threadIdx.x >> 6);
    const int t = threadIdx.x & 63;
    const int r = row[e];
    const int c = col[e];
    const float norm = dis[r] * dis[c];
    const float4 hv = *(const float4*)(h + (size_t)r * DD + t * 4);
    float* o = out + (size_t)c * DD + t * 4;
    atomicAdd(o + 0, hv.x * norm);
    atomicAdd(o + 1, hv.y * norm);
    atomicAdd(o + 2, hv.z * norm);
    atomicAdd(o + 3, hv.w * norm);
}

// ---------------- tanh ----------------
__global__ __launch_bounds__(256) void k_tanh(float* __restrict__ out) {
    size_t idx = (size_t)blockIdx.x * 256 + threadIdx.x;
    out[idx] = tanhf(out[idx]);
}

extern "C" void kernel_launch(void* const* d_in, const int* in_sizes, int n_in,
                              void* d_out, int out_size, void* d_ws, size_t ws_size,
                              hipStream_t stream) {
    const float* x   = (const float*)d_in[0];       // [NN, DD]
    const int*   ei  = (const int*)d_in[1];         // [2, EE] flat
    const float* W   = (const float*)d_in[2];       // [DD, DD] row-major [in][out]
    const float* bia = (const float*)d_in[3];       // [DD]
    float* out = (float*)d_out;                     // [NN, DD]

    const int* row = ei;        // messages flow row -> col
    const int* col = ei + EE;

    // workspace layout: deg/dis (200KB) | h (51.2MB)
    float* deg = (float*)d_ws;
    float* h   = (float*)((char*)d_ws + (1u << 20));

    k_deg_init <<<(NN + 255) / 256, 256, 0, stream>>>(deg);
    k_deg_count<<<(EE + 255) / 256, 256, 0, stream>>>(col, deg);
    k_deg_rsqrt<<<(NN + 255) / 256, 256, 0, stream>>>(deg);   // deg -> deg^{-1/2} in place

    k_gemm<<<dim3(NN / 16, DD / 64), 128, 0, stream>>>(x, W, h);

    k_init_out<<<NN, DD, 0, stream>>>(h, deg, bia, out);
    k_scatter <<<EE / 4, 256, 0, stream>>>(row, col, h, deg, out);
    k_tanh    <<<NN, 256, 0, stream>>>(out);
}